// SimpleAttentionKVCache_39247411150862
// MI455X (gfx1250) — compile-verified
//
#include <hip/hip_runtime.h>
#include <hip/hip_bf16.h>
#include <cstdint>
#include <cstddef>

typedef __attribute__((ext_vector_type(2))) float v2f;
typedef __attribute__((ext_vector_type(8))) float v8f;

#define B_      32
#define S_NEW   8
#define S_CACHE 4096
#define S_TOTAL 4104
#define D_      1024
#define M_ROWS  256      // B_*S_NEW
#define NCHUNK  9
#define CHUNK   512      // 8*512 + 8 = 4104

// ---------------------------------------------------------------------------
// Kernel 1: fused QKV projection GEMM, fp32 WMMA 16x16x4.
// grid = (16 ngroups, 16 mtiles, 3 weights), block = 128 (4 waves).
// Each wave owns one 16x16 output tile; A tile staged in padded LDS.
// ---------------------------------------------------------------------------
__global__ __launch_bounds__(128) void qkv_proj_kernel(
    const float* __restrict__ input,
    const float* __restrict__ Wk, const float* __restrict__ bk,
    const float* __restrict__ Wv, const float* __restrict__ bv,
    const float* __restrict__ Wq, const float* __restrict__ bq,
    float* __restrict__ out_keys, float* __restrict__ out_vals,
    float* __restrict__ q_ws)
{
    __shared__ float ldsA[16 * 1028];   // 16 rows, stride 1028 -> conflict-free
    const int tid   = threadIdx.x;
    const int mtile = blockIdx.y;
    const int wsel  = blockIdx.z;
    const float* W    = (wsel == 0) ? Wk : (wsel == 1) ? Wv : Wq;
    const float* bias = (wsel == 0) ? bk : (wsel == 1) ? bv : bq;

    // stage A tile: rows [mtile*16, mtile*16+16) of the flattened [256,1024] input
    const float4* in4 = (const float4*)(input + (size_t)mtile * 16 * D_);
    for (int idx = tid; idx < 16 * 256; idx += 128) {
        int row = idx >> 8, c4 = idx & 255;
        *(float4*)&ldsA[row * 1028 + c4 * 4] = in4[row * 256 + c4];
    }
    __syncthreads();

    const int wid  = tid >> 5, lane = tid & 31;
    const int lh   = lane & 15, hi = lane >> 4, koff = hi * 2;
    const int col  = blockIdx.x * 64 + wid * 16 + lh;

    v8f acc = {};
    const float* ldsRow = &ldsA[lh * 1028];
    #pragma unroll 4
    for (int k = 0; k < D_; k += 4) {
        v2f a = *(const v2f*)&ldsRow[k + koff];          // A[ M=lh ][ K=k+koff, +1 ]
        v2f b;
        b.x = W[(size_t)(k + koff)     * D_ + col];      // B[ K ][ N=col ]
        b.y = W[(size_t)(k + koff + 1) * D_ + col];
        acc = __builtin_amdgcn_wmma_f32_16x16x4_f32(false, a, false, b,
                                                    (short)0, acc, false, false);
    }

    const float bvv = bias[col];
    #pragma unroll
    for (int r = 0; r < 8; ++r) {
        const int m  = mtile * 16 + r + hi * 8;          // flat query row
        const int bb = m >> 3, ss = m & 7;
        const float v = acc[r] + bvv;
        if (wsel == 2) {
            q_ws[(size_t)m * D_ + col] = v;
        } else {
            float* dst = (wsel == 0) ? out_keys : out_vals;
            dst[((size_t)bb * S_TOTAL + S_CACHE + ss) * D_ + col] = v;
        }
    }
}

// ---------------------------------------------------------------------------
// Kernel 2: stream the two 512 MiB caches into the strided output layout.
// ---------------------------------------------------------------------------
__global__ __launch_bounds__(256) void cache_copy_kernel(
    const float4* __restrict__ kc, const float4* __restrict__ vc,
    float4* __restrict__ out_keys, float4* __restrict__ out_vals)
{
    const int tensor = blockIdx.y;
    const float4* src = tensor ? vc : kc;
    float4* dst       = tensor ? out_vals : out_keys;
    const size_t N4    = (size_t)B_ * S_CACHE * (D_ / 4);   // 33,554,432
    const size_t per_b = (size_t)S_CACHE * (D_ / 4);        // 2^20
    const size_t stride = (size_t)gridDim.x * blockDim.x;
    for (size_t i = (size_t)blockIdx.x * blockDim.x + threadIdx.x; i < N4; i += stride) {
        const size_t b   = i >> 20;
        const size_t rem = i & (per_b - 1);
        dst[b * ((size_t)S_TOTAL * (D_ / 4)) + rem] = src[i];
    }
}

// ---------------------------------------------------------------------------
// Kernel 3: flash-attention partials per (batch, key-chunk).
// grid = (NCHUNK, B_), block = 256 (8 waves).
// Per 128-key window: wave w computes one 16x16 QK^T tile (f32 WMMA),
// online softmax via LDS, then PV WMMA into its 128-col accumulator slice.
// ---------------------------------------------------------------------------
__global__ __launch_bounds__(256) void attn_partial_kernel(
    const float* __restrict__ q_ws,
    const float* __restrict__ keys, const float* __restrict__ values,
    float* __restrict__ part, float* __restrict__ mstat, float* __restrict__ lstat)
{
    __shared__ float ldsQ[16 * 1028];
    __shared__ float ldsS[16 * 132];
    __shared__ float rowM[16], rowL[16], rowFac[16];

    const int b = blockIdx.y, chunk = blockIdx.x;
    const int kb = chunk * CHUNK;
    const int ke = (kb + CHUNK < S_TOTAL) ? (kb + CHUNK) : S_TOTAL;
    const float* K = keys   + (size_t)b * S_TOTAL * D_;
    const float* V = values + (size_t)b * S_TOTAL * D_;
    const int tid = threadIdx.x;

    // stage Q (rows 0..7 real, 8..15 zero-padded)
    for (int idx = tid; idx < 16 * 256; idx += 256) {
        int row = idx >> 8, c4 = idx & 255;
        float4 v = {0.f, 0.f, 0.f, 0.f};
        if (row < S_NEW)
            v = *(const float4*)&q_ws[((size_t)(b * S_NEW + row)) * D_ + c4 * 4];
        *(float4*)&ldsQ[row * 1028 + c4 * 4] = v;
    }
    if (tid < 16) { rowM[tid] = -3.0e38f; rowL[tid] = 0.f; }
    __syncthreads();

    const int wid = tid >> 5, lane = tid & 31;
    const int lh = lane & 15, hi = lane >> 4, koff = hi * 2;
    const int n0 = wid * 128;                       // this wave's V-column slice

    v8f vzero = {};
    v8f acc[8];
    #pragma unroll
    for (int i = 0; i < 8; ++i) acc[i] = vzero;

    const float scale = 0.03125f;                   // 1/sqrt(1024)

    for (int t0 = kb; t0 < ke; t0 += 128) {
        // ---- scores: 16x16 tile, keys [tb, tb+16) ----
        const int tb = t0 + wid * 16;
        int kr = tb + lh; if (kr > S_TOTAL - 1) kr = S_TOTAL - 1;
        const float* Krow = K + (size_t)kr * D_;
        if (t0 + 128 < ke)                          // prefetch next window's K row
            __builtin_prefetch(K + (size_t)(kr + 128) * D_, 0, 0);

        v8f s = vzero;
        #pragma unroll 4
        for (int k = 0; k < D_; k += 4) {
            v2f a  = *(const v2f*)&ldsQ[lh * 1028 + k + koff];
            v2f bf = *(const v2f*)&Krow[k + koff];   // B[K][N=key lane]: contiguous pair
            s = __builtin_amdgcn_wmma_f32_16x16x4_f32(false, a, false, bf,
                                                      (short)0, s, false, false);
        }
        const bool valid = (tb + lh) < ke;
        #pragma unroll
        for (int r = 0; r < 8; ++r)
            ldsS[(r + hi * 8) * 132 + wid * 16 + lh] = valid ? s[r] * scale : -3.0e38f;
        __syncthreads();

        // ---- online softmax update (one thread per query row) ----
        if (tid < 16) {
            const int r = tid;
            float mo = rowM[r], mx = mo;
            int cnt = ke - t0; if (cnt > 128) cnt = 128;
            for (int j = 0; j < cnt; ++j) mx = fmaxf(mx, ldsS[r * 132 + j]);
            const float fac = __expf(mo - mx);
            float sum = rowL[r] * fac;
            for (int j = 0; j < 128; ++j) {
                const float p = (j < cnt) ? __expf(ldsS[r * 132 + j] - mx) : 0.f;
                ldsS[r * 132 + j] = p;
                sum += p;
            }
            rowM[r] = mx; rowL[r] = sum; rowFac[r] = fac;
        }
        __syncthreads();

        // ---- rescale accumulators ----
        float f[8];
        #pragma unroll
        for (int r = 0; r < 8; ++r) f[r] = rowFac[r + hi * 8];
        #pragma unroll
        for (int nt = 0; nt < 8; ++nt) {
            #pragma unroll
            for (int r = 0; r < 8; ++r) acc[nt][r] *= f[r];
        }

        // ---- PV: acc += P[16x128] * V[128x128-slice] ----
        for (int kk = 0; kk < 128; kk += 4) {
            v2f a = *(const v2f*)&ldsS[lh * 132 + kk + koff];   // P[M=lh][K=kk+koff,+1]
            int vr0 = t0 + kk + koff;
            if (vr0 > S_TOTAL - 2) vr0 = S_TOTAL - 2;            // p==0 there anyway
            const float* Vr0 = V + (size_t)vr0 * D_;
            #pragma unroll
            for (int nt = 0; nt < 8; ++nt) {
                const int cc = n0 + nt * 16 + lh;
                v2f bf; bf.x = Vr0[cc]; bf.y = Vr0[D_ + cc];
                acc[nt] = __builtin_amdgcn_wmma_f32_16x16x4_f32(false, a, false, bf,
                                                                (short)0, acc[nt], false, false);
            }
        }
        __syncthreads();   // before next window overwrites ldsS
    }

    // ---- write partial O (rows 0..7 live in the hi==0 half-wave) and stats ----
    const size_t pbase = ((size_t)(b * NCHUNK + chunk)) * S_NEW * D_;
    if (hi == 0) {
        #pragma unroll
        for (int nt = 0; nt < 8; ++nt) {
            #pragma unroll
            for (int r = 0; r < 8; ++r)
                part[pbase + (size_t)r * D_ + n0 + nt * 16 + lh] = acc[nt][r];
        }
    }
    if (tid < S_NEW) {
        mstat[(b * NCHUNK + chunk) * S_NEW + tid] = rowM[tid];
        lstat[(b * NCHUNK + chunk) * S_NEW + tid] = rowL[tid];
    }
}

// ---------------------------------------------------------------------------
// Kernel 4: split-softmax reduction across NCHUNK partials.
// grid = 256 blocks (one per (b,s)), block = 256.
// ---------------------------------------------------------------------------
__global__ __launch_bounds__(256) void attn_reduce_kernel(
    const float* __restrict__ part, const float* __restrict__ mstat,
    const float* __restrict__ lstat, float* __restrict__ out)
{
    const int bs = blockIdx.x;
    const int b = bs >> 3, s = bs & 7;
    float m[NCHUNK], l[NCHUNK], w[NCHUNK];
    float M = -3.0e38f;
    for (int c = 0; c < NCHUNK; ++c) {
        m[c] = mstat[(b * NCHUNK + c) * S_NEW + s];
        l[c] = lstat[(b * NCHUNK + c) * S_NEW + s];
        M = fmaxf(M, m[c]);
    }
    float L = 0.f;
    for (int c = 0; c < NCHUNK; ++c) { w[c] = __expf(m[c] - M); L += l[c] * w[c]; }
    const float invL = 1.0f / L;
    for (int col = threadIdx.x; col < D_; col += 256) {
        float a = 0.f;
        for (int c = 0; c < NCHUNK; ++c)
            a += part[((size_t)(b * NCHUNK + c) * S_NEW + s) * D_ + col] * w[c];
        out[((size_t)(b * S_NEW) + s) * D_ + col] = a * invL;
    }
}

// ---------------------------------------------------------------------------
extern "C" void kernel_launch(void* const* d_in, const int* in_sizes, int n_in,
                              void* d_out, int out_size, void* d_ws, size_t ws_size,
                              hipStream_t stream) {
    const float* input = (const float*)d_in[0];
    const float* kc    = (const float*)d_in[1];
    const float* vc    = (const float*)d_in[2];
    const float* Wk    = (const float*)d_in[3];
    const float* bk    = (const float*)d_in[4];
    const float* Wv    = (const float*)d_in[5];
    const float* bv    = (const float*)d_in[6];
    const float* Wq    = (const float*)d_in[7];
    const float* bq    = (const float*)d_in[8];

    float* out      = (float*)d_out;                              // [B,8,1024]
    float* out_keys = out + (size_t)B_ * S_NEW * D_;              // [B,4104,1024]
    float* out_vals = out_keys + (size_t)B_ * S_TOTAL * D_;       // [B,4104,1024]

    float* ws    = (float*)d_ws;
    float* q_ws  = ws;                                            // 256*1024
    float* part  = q_ws + (size_t)M_ROWS * D_;                    // B*9*8*1024
    float* mstat = part + (size_t)B_ * NCHUNK * S_NEW * D_;       // B*9*8
    float* lstat = mstat + (size_t)B_ * NCHUNK * S_NEW;           // B*9*8

    qkv_proj_kernel<<<dim3(16, 16, 3), 128, 0, stream>>>(
        input, Wk, bk, Wv, bv, Wq, bq, out_keys, out_vals, q_ws);

    cache_copy_kernel<<<dim3(2048, 2), 256, 0, stream>>>(
        (const float4*)kc, (const float4*)vc, (float4*)out_keys, (float4*)out_vals);

    attn_partial_kernel<<<dim3(NCHUNK, B_), 256, 0, stream>>>(
        q_ws, out_keys, out_vals, part, mstat, lstat);

    attn_reduce_kernel<<<256, 256, 0, stream>>>(part, mstat, lstat, out);
}